// DS2_62466004353281
// MI455X (gfx1250) — compile-verified
//
#include <hip/hip_runtime.h>

// MI455X / gfx1250 — wave32, WMMA.
// out[p, i, c] = in[p, i] * u[i, c],  u = beta^2 row-normalized, D=64, C=16.
// Memory-bound: 33.5 MB read once + 536.9 MB written once => ~24.5 us roofline @ 23.3 TB/s.
// Broadcast multiply realized as rank-1 V_WMMA_F32_16X16X4_F32 per (16-pixel group, i);
// compute is ~1 GFLOP of matrix work, fully hidden under the store stream.
// All stores are non-temporal (write-once 537 MB must not thrash the 192 MB L2).

typedef __attribute__((ext_vector_type(2))) float v2f;
typedef __attribute__((ext_vector_type(4))) float v4f;
typedef __attribute__((ext_vector_type(8))) float v8f;

#define D_DIM 64
#define C_DIM 16
#define PIX_PER_BLOCK 128          // 8 waves x 16 pixels
#define LDS_STRIDE 68              // 64 + 4 pad: conflict-free column reads, rows stay 16B aligned

__global__ __launch_bounds__(256) void mass_proto_wmma_kernel(
    const float* __restrict__ in,    // [Npix, 64]
    const float* __restrict__ beta,  // [64, 16]
    float* __restrict__ out)         // [Npix, 64, 16]
{
    __shared__ __align__(16) float s_in[PIX_PER_BLOCK * LDS_STRIDE];
    __shared__ __align__(16) float s_u[D_DIM * C_DIM];
    __shared__ __align__(16) float s_zero[4];   // broadcast-zero slot for lanes 16..31 (A's K=2 half)

    const int t = threadIdx.x;

    // ---- u[i,c] = beta[i,c]^2 / sum_c beta[i,c]^2  (one row per thread, t<64) ----
    if (t < D_DIM) {
        const float* bp = beta + t * C_DIM;
        float row[C_DIM];
        float s = 0.0f;
#pragma unroll
        for (int c = 0; c < C_DIM; ++c) { float v = bp[c]; row[c] = v * v; s += row[c]; }
        const float inv = 1.0f / s;
#pragma unroll
        for (int c = 0; c < C_DIM; ++c) s_u[t * C_DIM + c] = row[c] * inv;
    }
    if (t == D_DIM) { s_zero[0] = 0.0f; s_zero[1] = 0.0f; s_zero[2] = 0.0f; s_zero[3] = 0.0f; }

    // ---- stage 128x64 input tile into LDS: coalesced b128 NT loads, each element read once ----
    const size_t pixBlock = (size_t)blockIdx.x * PIX_PER_BLOCK;
    const v4f* in4 = (const v4f*)(in + pixBlock * D_DIM);   // 16 float4 per pixel row
#pragma unroll
    for (int k = 0; k < 8; ++k) {
        const int idx = t + k * 256;          // 0..2047
        const int r  = idx >> 4;              // pixel row in tile
        const int c4 = idx & 15;              // float4 column
        v4f v = __builtin_nontemporal_load(&in4[r * 16 + c4]);
        *(v4f*)&s_in[r * LDS_STRIDE + c4 * 4] = v;
    }
    __syncthreads();

    // ---- per wave: 16 pixels; loop i over D: rank-1 WMMA produces 16x16 (pixel x c) tile ----
    const int lane = t & 31;                  // wave32
    const int wave = t >> 5;                  // 0..7
    const int rowBase = wave * 16;            // first tile row for this wave
    const int hi = lane >> 4;                 // 0: M=0..7 rows, 1: M=8..15 rows
    const int lc = lane & 15;                 // N (= c) index

    // A source: lanes 0-15 walk their pixel row (K=0 column); lanes 16-31 (A's K=2
    // slot) read a fixed zeroed LDS word — unconditional ds_load, no EXEC masking,
    // broadcast (no bank conflict) for the upper half-wave.
    const float* aptr = (lane < 16) ? &s_in[(rowBase + lane) * LDS_STRIDE] : &s_zero[0];
    const int    astep = (lane < 16) ? 1 : 0;

    float* outBase = out + (pixBlock + (size_t)rowBase) * (D_DIM * C_DIM)
                   + (size_t)hi * 8 * (D_DIM * C_DIM) + lc;

#pragma unroll 4
    for (int i = 0; i < D_DIM; ++i) {
        // A (16x4 f32): K=0 column = input pixels; K=1..3 zero.
        const float a0 = *aptr;
        v2f A = { a0, 0.0f };

        // B (4x16 f32): only K=0 row can contribute (A zero elsewhere); second
        // VGPR stays a constant 0 so no per-iteration replication mov is needed.
        const float uv = s_u[i * C_DIM + lc];
        v2f Bm = { uv, 0.0f };

        v8f Cm = {};  // zero accumulator
        // 8 args: (neg_a, A, neg_b, B, c_mod, C, reuse_a, reuse_b)
        v8f Dm = __builtin_amdgcn_wmma_f32_16x16x4_f32(
            false, A, false, Bm, (short)0, Cm, false, false);

        // D layout: VGPR r, lanes 0-15 -> (M=r, N=lane); lanes 16-31 -> (M=r+8, N=lane-16).
        // out addr for (pixel M, i, c=N) = outBase + r*1024 (+ i*16 folded into pointer).
        float* o = outBase + (size_t)i * C_DIM;
#pragma unroll
        for (int r = 0; r < 8; ++r)
            __builtin_nontemporal_store(Dm[r], o + (size_t)r * (D_DIM * C_DIM));

        aptr += astep;
    }
}

extern "C" void kernel_launch(void* const* d_in, const int* in_sizes, int n_in,
                              void* d_out, int out_size, void* d_ws, size_t ws_size,
                              hipStream_t stream) {
    const float* in   = (const float*)d_in[0];   // [B*H*W*D] f32
    const float* beta = (const float*)d_in[1];   // [D*C] f32
    float* out        = (float*)d_out;           // [B*H*W*D*C] f32

    const int n_pixels = in_sizes[0] / D_DIM;            // 131072 for reference shapes
    const int grid     = n_pixels / PIX_PER_BLOCK;       // 1024 (exact)

    mass_proto_wmma_kernel<<<grid, 256, 0, stream>>>(in, beta, out);
}